// FP4RoundingPerturb_21947282882569
// MI455X (gfx1250) — compile-verified
//
#include <hip/hip_runtime.h>

// FP4 E2M1 fake-quant with perturbed bounds + soft ordinal weights.
// MI455X reasoning: ~370MB traffic (268MB of it the [...,8] weights output)
// => store-bandwidth bound (~16us @ 23.3TB/s). Strategy:
//  - async global->LDS staging of x (ASYNCcnt path), overlap with uniform loads
//  - wave32-native block reduction: one 32-elem quant block = 4 lanes x 8 elems,
//    amax via two shfl_xor hops
//  - b128 non-temporal stores for all bulk outputs (write-once, bypass/rinse L2)
//  - v_exp_f32 / v_rcp_f32 / v_log_f32 fast transcendentals for the 7 sigmoids

typedef float v4f  __attribute__((ext_vector_type(4)));
typedef float v4fu __attribute__((ext_vector_type(4), aligned(4)));  // weights region is +28B misaligned

#define K_SIG 14.426950408889634f   // log2(e) / TAU, TAU = 0.1

__global__ __launch_bounds__(256) void fp4_round_perturb_kernel(
    const float* __restrict__ x,
    const float* __restrict__ delta_raw,
    const float* __restrict__ bounds_base,
    const float* __restrict__ values_table,
    float* __restrict__ out,
    long long N)
{
    __shared__ __align__(16) float s_x[2048];   // 256 threads * 8 floats

    const int tid = threadIdx.x;
    const long long tile_base = (long long)blockIdx.x * 2048;
    const long long g = tile_base + (long long)tid * 8;   // first of 8 contiguous elems
    if (g + 8 > N) return;   // N % 2048 == 0 for the given shapes

    // ---- stage 1: async global->LDS copy of this lane's 32 bytes (2 x b128) ----
    const float* tile = x + tile_base;
    unsigned voff     = (unsigned)tid * 32u;
    unsigned lds_addr = (unsigned)(size_t)&s_x[tid * 8];  // low 32 bits = LDS byte offset
    asm volatile(
        "global_load_async_to_lds_b128 %0, %1, %2\n\t"
        "global_load_async_to_lds_b128 %0, %1, %2 offset:16"
        :: "v"(lds_addr), "v"(voff), "s"(tile) : "memory");

    // ---- overlap the async copy with uniform parameter setup ----
    const float shift = 0.5f * tanhf(delta_raw[0]);
    float sb[7];
#pragma unroll
    for (int i = 0; i < 7; ++i) sb[i] = bounds_base[i] + shift;
    float vt[8];
#pragma unroll
    for (int i = 0; i < 8; ++i) vt[i] = values_table[i];

    // output layout (all float): y[N] | codes[N] | scales[N/32] | bounds[7] | weights[8N]
    float* y_out = out;
    float* c_out = out + N;
    float* s_out = out + 2 * N;
    float* b_out = s_out + (N >> 5);
    float* w_out = b_out + 7;

    if (blockIdx.x == 0 && tid < 7) b_out[tid] = sb[tid];

    asm volatile("s_wait_asynccnt 0" ::: "memory");

    const v4f a = *(const v4f*)(&s_x[tid * 8]);
    const v4f b = *(const v4f*)(&s_x[tid * 8 + 4]);

    // ---- per-block (32 elems = 4 lanes) amax via wave32 shuffles ----
    float m = fmaxf(fmaxf(fmaxf(fabsf(a.x), fabsf(a.y)), fmaxf(fabsf(a.z), fabsf(a.w))),
                    fmaxf(fmaxf(fabsf(b.x), fabsf(b.y)), fmaxf(fabsf(b.z), fabsf(b.w))));
    m = fmaxf(m, __shfl_xor(m, 1, 32));
    m = fmaxf(m, __shfl_xor(m, 2, 32));

    // e8m0 power-of-two scale: e = ceil(max(log2(amax/6), -127))
    const float e      = ceilf(fmaxf(__builtin_amdgcn_logf(m * (1.0f / 6.0f)), -127.0f));
    const float sf     = __builtin_amdgcn_exp2f(e);
    const float inv_sf = __builtin_amdgcn_exp2f(-e);

    if ((tid & 3) == 0)
        __builtin_nontemporal_store(e + 127.0f, s_out + (g >> 5));

    const float xv[8] = {a.x, a.y, a.z, a.w, b.x, b.y, b.z, b.w};
    float yv[8], cv[8];

#pragma unroll
    for (int j = 0; j < 8; ++j) {
        const float xs = xv[j] * inv_sf;
        const float xa = fabsf(xs);

        // 7 boundary sigmoids: p = 1 / (1 + exp2((sb - xa) * log2e/tau))
        float p[7];
#pragma unroll
        for (int k = 0; k < 7; ++k)
            p[k] = __builtin_amdgcn_rcpf(1.0f + __builtin_amdgcn_exp2f((sb[k] - xa) * K_SIG));

        float w[8];
        w[0] = fmaxf(1.0f - p[0], 0.0f);
#pragma unroll
        for (int k = 1; k < 7; ++k) w[k] = fmaxf(p[k - 1] - p[k], 0.0f);
        w[7] = fmaxf(p[6], 0.0f);

        float ws = w[0];
#pragma unroll
        for (int k = 1; k < 8; ++k) ws += w[k];
        const float wn = __builtin_amdgcn_rcpf(ws + 1e-8f);

        const v4fu wlo = {w[0] * wn, w[1] * wn, w[2] * wn, w[3] * wn};
        const v4fu whi = {w[4] * wn, w[5] * wn, w[6] * wn, w[7] * wn};
        float* wp = w_out + (size_t)(g + j) * 8;
        __builtin_nontemporal_store(wlo, (v4fu*)wp);
        __builtin_nontemporal_store(whi, (v4fu*)(wp + 4));

        // hard ordinal code + hard dequant value (select chain, no dyn-index)
        int   ord = 0;
        float ah  = vt[0];
#pragma unroll
        for (int k = 0; k < 7; ++k) {
            const bool gt = xa > sb[k];
            ord += gt ? 1 : 0;
            ah = gt ? vt[k + 1] : ah;
        }
        // forward pass of the straight-through mix == hard value
        yv[j] = ((xs >= 0.0f) ? ah : -ah) * sf;
        cv[j] = (float)(((xs < 0.0f) ? 8 : 0) | ord);
    }

    const v4f y0 = {yv[0], yv[1], yv[2], yv[3]};
    const v4f y1 = {yv[4], yv[5], yv[6], yv[7]};
    const v4f c0 = {cv[0], cv[1], cv[2], cv[3]};
    const v4f c1 = {cv[4], cv[5], cv[6], cv[7]};
    __builtin_nontemporal_store(y0, (v4f*)(y_out + g));
    __builtin_nontemporal_store(y1, (v4f*)(y_out + g + 4));
    __builtin_nontemporal_store(c0, (v4f*)(c_out + g));
    __builtin_nontemporal_store(c1, (v4f*)(c_out + g + 4));
}

extern "C" void kernel_launch(void* const* d_in, const int* in_sizes, int n_in,
                              void* d_out, int out_size, void* d_ws, size_t ws_size,
                              hipStream_t stream) {
    const float* x      = (const float*)d_in[0];
    const float* delta  = (const float*)d_in[1];
    const float* bounds = (const float*)d_in[2];
    const float* values = (const float*)d_in[3];
    float* out = (float*)d_out;

    const long long N = (long long)in_sizes[0];       // 2048*4096
    const int grid = (int)((N + 2047) / 2048);        // 2048 elems per 256-thread block
    fp4_round_perturb_kernel<<<grid, 256, 0, stream>>>(x, delta, bounds, values, out, N);
}